// SpatialInteractionLayer_67894843015270
// MI455X (gfx1250) — compile-verified
//
#include <hip/hip_runtime.h>
#include <hip/hip_bf16.h>

#define BB 8
#define NN 384
#define DM 512
#define HH 8
#define DH 64
#define SCALE 0.125f

typedef float v2f __attribute__((ext_vector_type(2)));
typedef float v8f __attribute__((ext_vector_type(8)));

__device__ __forceinline__ v8f wmma_f32(v2f a, v2f b, v8f c) {
  // D = A(16x4, f32) x B(4x16, f32) + C(16x16, f32)  -> v_wmma_f32_16x16x4_f32
  return __builtin_amdgcn_wmma_f32_16x16x4_f32(false, a, false, b, (short)0, c,
                                               false, false);
}

// ---------------------------------------------------------------------------
// Kernel 1: QKV projection.  (3072 x 512) @ (512 x 1536) + bqkv
// Block = 256 threads = 8 waves; block tile 64(M) x 64(N); each wave 16x32.
// Scatter result into q/k/v with [b][h][n][d] layout.
// ---------------------------------------------------------------------------
__global__ __launch_bounds__(256) void qkv_kernel(
    const float* __restrict__ x, const float* __restrict__ Wqkv,
    const float* __restrict__ bqkv, float* __restrict__ q_ws,
    float* __restrict__ k_ws, float* __restrict__ v_ws) {
  const int lane = threadIdx.x & 31;
  const int wave = threadIdx.x >> 5;
  const int m16 = lane & 15;
  const int kh = lane >> 4;
  const int mi = wave & 3;   // 4 m-tiles of 16
  const int ni2 = wave >> 2; // 2 n-groups of 32

  const int row0 = blockIdx.x * 64;
  const int col0 = blockIdx.y * 64;

  const int row = row0 + mi * 16 + m16;
  const int colA = col0 + ni2 * 32 + m16;

  const float* xrow = x + (size_t)row * DM;

  v8f acc0 = {0.f, 0.f, 0.f, 0.f, 0.f, 0.f, 0.f, 0.f};
  v8f acc1 = {0.f, 0.f, 0.f, 0.f, 0.f, 0.f, 0.f, 0.f};

#pragma unroll 4
  for (int k = 0; k < DM; k += 4) {
    const int kr = k + 2 * kh;
    v2f a = *(const v2f*)(xrow + kr);
    v2f b0, b1;
    b0.x = Wqkv[(size_t)kr * 1536 + colA];
    b0.y = Wqkv[(size_t)(kr + 1) * 1536 + colA];
    b1.x = Wqkv[(size_t)kr * 1536 + colA + 16];
    b1.y = Wqkv[(size_t)(kr + 1) * 1536 + colA + 16];
    acc0 = wmma_f32(a, b0, acc0);
    acc1 = wmma_f32(a, b1, acc1);
  }

  // Epilogue: tile rows never straddle a batch boundary (384 % 16 == 0),
  // so batch / sequence indices are tile-uniform -> hoist all divisions.
  const int grow_base = row0 + mi * 16;
  const int gb = grow_base / NN;
  const int gn0 = (grow_base % NN) + kh * 8;

#pragma unroll
  for (int t = 0; t < 2; ++t) {
    const int col = colA + t * 16;
    const int s = col >> 9;  // 0=q 1=k 2=v
    const int c2 = col & 511;
    const int hh = c2 >> 6;
    const int d = c2 & 63;
    float* dst = (s == 0) ? q_ws : (s == 1) ? k_ws : v_ws;
    float* p = dst + ((((size_t)(gb * HH + hh) * NN) + gn0) * DH + d);
    const float bz = bqkv[col];
#pragma unroll
    for (int e = 0; e < 8; ++e) {
      p[(size_t)e * DH] = ((t == 0) ? acc0[e] : acc1[e]) + bz;
    }
  }
}

// ---------------------------------------------------------------------------
// Kernel 2: relative-position bias MLP.
// bias[b][i][j][h] = (relu(rel[b,i,j,:] @ W1 + b1) @ W2 + b2)[h]
// One thread per (b,i,j); weights staged in LDS.
// ---------------------------------------------------------------------------
__global__ __launch_bounds__(256) void bias_mlp_kernel(
    const float* __restrict__ rel, const float* __restrict__ W1,
    const float* __restrict__ b1, const float* __restrict__ W2,
    const float* __restrict__ b2, float* __restrict__ bias_ws) {
  __shared__ float sW1[128];
  __shared__ float sb1[64];
  __shared__ float sW2[512];
  __shared__ float sb2[8];
  const int t = threadIdx.x;
  if (t < 128) sW1[t] = W1[t];
  if (t < 64) sb1[t] = b1[t];
  if (t < 8) sb2[t] = b2[t];
  for (int i = t; i < 512; i += 256) sW2[i] = W2[i];
  __syncthreads();

  const size_t p = (size_t)blockIdx.x * 256 + t;
  if (p >= (size_t)BB * NN * NN) return;
  const float rx = rel[p * 2 + 0];
  const float ry = rel[p * 2 + 1];

  float acc[8];
#pragma unroll
  for (int e = 0; e < 8; ++e) acc[e] = sb2[e];

#pragma unroll 8
  for (int o = 0; o < 64; ++o) {
    const float hid =
        fmaxf(fmaf(rx, sW1[o], fmaf(ry, sW1[64 + o], sb1[o])), 0.f);
#pragma unroll
    for (int e = 0; e < 8; ++e) acc[e] = fmaf(hid, sW2[o * 8 + e], acc[e]);
  }
#pragma unroll
  for (int e = 0; e < 8; ++e) bias_ws[p * 8 + e] = acc[e];
}

// ---------------------------------------------------------------------------
// Kernel 3: fused attention for one (b, h, 32-query tile).
// S = Q Kt * SCALE + bias, mask, softmax (LDS), O = P V.  All matmuls WMMA f32.
// ---------------------------------------------------------------------------
#define TQ 32
__global__ __launch_bounds__(256) void attn_kernel(
    const float* __restrict__ q_ws, const float* __restrict__ k_ws,
    const float* __restrict__ v_ws, const float* __restrict__ bias_ws,
    const unsigned char* __restrict__ mask, float* __restrict__ ao_ws) {
  __shared__ float S[TQ * NN];  // 32*384*4 = 48 KB

  const int b = blockIdx.z;
  const int h = blockIdx.y;
  const int q0 = blockIdx.x * TQ;

  const float* qb = q_ws + (((size_t)(b * HH + h) * NN) + q0) * DH;
  const float* kb = k_ws + ((size_t)(b * HH + h) * NN) * DH;
  const float* vb = v_ws + ((size_t)(b * HH + h) * NN) * DH;

  const int lane = threadIdx.x & 31;
  const int wave = threadIdx.x >> 5;
  const int m16 = lane & 15;
  const int kh = lane >> 4;

  // ---- Phase 1: S tiles.  2 m-tiles x 8 groups of 3 n-tiles = 16 groups,
  // exactly 2 per wave.  One A fragment feeds 3 WMMAs.
  for (int t = wave; t < 16; t += 8) {
    const int mi = t >> 3;        // 0..1
    const int njg = t & 7;        // group of 3 n-tiles: cols njg*48 .. +47
    v8f acc[3];
#pragma unroll
    for (int u = 0; u < 3; ++u)
      acc[u] = (v8f){0.f, 0.f, 0.f, 0.f, 0.f, 0.f, 0.f, 0.f};

    const float* arow = qb + (mi * 16 + m16) * DH;          // Q row
    const float* brow = kb + (njg * 48 + m16) * DH;         // K rows (B cols)
#pragma unroll 4
    for (int kk = 0; kk < DH; kk += 4) {
      v2f a = *(const v2f*)(arow + kk + 2 * kh);
#pragma unroll
      for (int u = 0; u < 3; ++u) {
        v2f bv = *(const v2f*)(brow + (u * 16) * DH + kk + 2 * kh);
        acc[u] = wmma_f32(a, bv, acc[u]);
      }
    }
#pragma unroll
    for (int u = 0; u < 3; ++u) {
      const int cn = njg * 48 + u * 16 + m16;
      const bool mok = mask[b * NN + cn] != 0;
#pragma unroll
      for (int e = 0; e < 8; ++e) {
        const int r = mi * 16 + kh * 8 + e;
        const float bz =
            bias_ws[(((size_t)(b * NN + q0 + r)) * NN + cn) * HH + h];
        const float val = fmaf(acc[u][e], SCALE, bz);
        S[r * NN + cn] = mok ? val : -__builtin_inff();
      }
    }
  }
  __syncthreads();

  // ---- Phase 2: row softmax (each wave handles 4 rows, lanes stride cols)
  for (int r = wave; r < TQ; r += 8) {
    float* row = S + r * NN;
    float mx = -__builtin_inff();
    for (int j = lane; j < NN; j += 32) mx = fmaxf(mx, row[j]);
#pragma unroll
    for (int off = 16; off > 0; off >>= 1) mx = fmaxf(mx, __shfl_xor(mx, off));
    float sum = 0.f;
    for (int j = lane; j < NN; j += 32) {
      const float e = __expf(row[j] - mx);
      row[j] = e;
      sum += e;
    }
#pragma unroll
    for (int off = 16; off > 0; off >>= 1) sum += __shfl_xor(sum, off);
    const float inv = 1.f / sum;
    for (int j = lane; j < NN; j += 32) row[j] *= inv;
  }
  __syncthreads();

  // ---- Phase 3: O = P @ V   (2 m-tiles x 4 n-tiles = 8 tiles, 1 per wave)
  for (int t = wave; t < 8; t += 8) {
    const int mi = t >> 2;
    const int nj = t & 3;
    v8f acc = {0.f, 0.f, 0.f, 0.f, 0.f, 0.f, 0.f, 0.f};
    const float* arow = S + (mi * 16 + m16) * NN;  // P row (LDS)
#pragma unroll 4
    for (int kk = 0; kk < NN; kk += 4) {
      const int kr = kk + 2 * kh;
      v2f a = *(const v2f*)(arow + kr);
      v2f bv;
      bv.x = vb[(size_t)kr * DH + nj * 16 + m16];
      bv.y = vb[(size_t)(kr + 1) * DH + nj * 16 + m16];
      acc = wmma_f32(a, bv, acc);
    }
#pragma unroll
    for (int e = 0; e < 8; ++e) {
      const int r = mi * 16 + kh * 8 + e;
      const int d = nj * 16 + m16;
      ao_ws[((size_t)(b * NN + q0 + r)) * DM + h * DH + d] = acc[e];
    }
  }
}

// ---------------------------------------------------------------------------
// Kernel 4: output projection.  (3072 x 512) @ (512 x 512) + bproj -> d_out
// ---------------------------------------------------------------------------
__global__ __launch_bounds__(256) void proj_kernel(
    const float* __restrict__ ao, const float* __restrict__ Wproj,
    const float* __restrict__ bproj, float* __restrict__ out) {
  const int lane = threadIdx.x & 31;
  const int wave = threadIdx.x >> 5;
  const int m16 = lane & 15;
  const int kh = lane >> 4;
  const int mi = wave & 3;
  const int ni2 = wave >> 2;

  const int row0 = blockIdx.x * 64;
  const int col0 = blockIdx.y * 64;
  const int row = row0 + mi * 16 + m16;
  const int colA = col0 + ni2 * 32 + m16;

  const float* arow = ao + (size_t)row * DM;

  v8f acc0 = {0.f, 0.f, 0.f, 0.f, 0.f, 0.f, 0.f, 0.f};
  v8f acc1 = {0.f, 0.f, 0.f, 0.f, 0.f, 0.f, 0.f, 0.f};

#pragma unroll 4
  for (int k = 0; k < DM; k += 4) {
    const int kr = k + 2 * kh;
    v2f a = *(const v2f*)(arow + kr);
    v2f b0, b1;
    b0.x = Wproj[(size_t)kr * DM + colA];
    b0.y = Wproj[(size_t)(kr + 1) * DM + colA];
    b1.x = Wproj[(size_t)kr * DM + colA + 16];
    b1.y = Wproj[(size_t)(kr + 1) * DM + colA + 16];
    acc0 = wmma_f32(a, b0, acc0);
    acc1 = wmma_f32(a, b1, acc1);
  }

  const int grow0 = row0 + mi * 16 + kh * 8;
#pragma unroll
  for (int t = 0; t < 2; ++t) {
    const int col = colA + t * 16;
    const float bz = bproj[col];
#pragma unroll
    for (int e = 0; e < 8; ++e) {
      out[(size_t)(grow0 + e) * DM + col] = ((t == 0) ? acc0[e] : acc1[e]) + bz;
    }
  }
}

// ---------------------------------------------------------------------------
extern "C" void kernel_launch(void* const* d_in, const int* in_sizes, int n_in,
                              void* d_out, int out_size, void* d_ws,
                              size_t ws_size, hipStream_t stream) {
  const float* x = (const float*)d_in[0];
  const float* rel = (const float*)d_in[1];
  const unsigned char* agent_mask = (const unsigned char*)d_in[2];
  const float* Wqkv = (const float*)d_in[3];
  const float* bqkv = (const float*)d_in[4];
  const float* Wproj = (const float*)d_in[5];
  const float* bproj = (const float*)d_in[6];
  const float* W1 = (const float*)d_in[7];
  const float* b1 = (const float*)d_in[8];
  const float* W2 = (const float*)d_in[9];
  const float* b2 = (const float*)d_in[10];

  float* ws = (float*)d_ws;
  const size_t QKV_ELEMS = (size_t)BB * HH * NN * DH;  // 1,572,864
  float* q_ws = ws;
  float* k_ws = q_ws + QKV_ELEMS;
  float* v_ws = k_ws + QKV_ELEMS;
  float* ao_ws = v_ws + QKV_ELEMS;
  float* bias_ws = ao_ws + QKV_ELEMS;  // B*N*N*H = 9,437,184 floats

  // 1) QKV projection (WMMA f32)
  qkv_kernel<<<dim3(48, 24), 256, 0, stream>>>(x, Wqkv, bqkv, q_ws, k_ws, v_ws);

  // 2) relative-position bias MLP (VALU)
  bias_mlp_kernel<<<dim3((BB * NN * NN) / 256), 256, 0, stream>>>(
      rel, W1, b1, W2, b2, bias_ws);

  // 3) fused attention (WMMA f32 + LDS softmax)
  attn_kernel<<<dim3(NN / TQ, HH, BB), 256, 0, stream>>>(
      q_ws, k_ws, v_ws, bias_ws, agent_mask, ao_ws);

  // 4) output projection (WMMA f32)
  proj_kernel<<<dim3(48, 8), 256, 0, stream>>>(ao_ws, Wproj, bproj,
                                               (float*)d_out);
}